// YOLOLayer_53532472377900
// MI455X (gfx1250) — compile-verified
//
#include <hip/hip_runtime.h>

typedef __attribute__((ext_vector_type(2))) float v2f;
typedef __attribute__((ext_vector_type(8))) float v8f;

#define CIN     256
#define HW      16384      // 128*128
#define WIMG    128
#define WSTRIDE 260        // LDS row stride (floats): 4n bank pattern, conflict-free b64 reads

__device__ __forceinline__ float sigmoidf(float v) {
    return 1.0f / (1.0f + __expf(-v));
}

__launch_bounds__(256, 4)
__global__ void yolo_wmma_kernel(const float* __restrict__ z,
                                 const float* __restrict__ wgt,
                                 const float* __restrict__ bias,
                                 float* __restrict__ out) {
    __shared__ float Ws[32 * WSTRIDE];   // Ws[o][c] = Wt padded: outs 18..31 zero

    const int tid = threadIdx.x;
    // Stage W^T (as Ws[o][c]) into LDS; pad unused out-channels with zeros.
    for (int i = tid; i < 32 * CIN; i += 256) {
        const int o = i >> 8;
        const int c = i & 255;
        Ws[o * WSTRIDE + c] = (o < 18) ? wgt[o * CIN + c] : 0.0f;
    }
    __syncthreads();

    const int lane   = tid & 31;
    const int waveId = tid >> 5;
    const int tile   = blockIdx.x * 8 + waveId;   // 16384 tiles total
    const int b      = tile >> 10;                // 1024 tiles per batch image
    const int rem    = tile & 1023;
    const int h      = rem >> 3;                  // row
    const int w0     = (rem & 7) << 4;            // 16-pixel column base
    const int pix    = rem << 4;                  // == h*128 + w0

    const int g = lane >> 4;                      // half-wave select (K pair)
    const int n = lane & 15;                      // pixel (A) / out-channel (B/C) index

    // A operand: z[channel = cbase + 2g + j][pixel = pix + n]
    const float* aptr = z + (size_t)b * (CIN * HW) + (size_t)(2 * g) * HW + (pix + n);
    // B operand: Ws[out][channel], lane reads channels (cbase+2g, cbase+2g+1) contiguously
    const float* bp0 = &Ws[n * WSTRIDE + 2 * g];
    const float* bp1 = &Ws[(n + 16) * WSTRIDE + 2 * g];

    v8f acc0 = {};   // outs 0..15  x pixels 0..15
    v8f acc1 = {};   // outs 16..31 x pixels 0..15 (only 16,17 valid)

#pragma unroll 4
    for (int k = 0; k < CIN; k += 4) {
        v2f a;
        a.x = aptr[0];        // K = 2g
        a.y = aptr[HW];       // K = 2g+1
        v2f b0 = *(const v2f*)bp0;   // 8B-aligned: (n*260 + 2g + k) is even
        v2f b1 = *(const v2f*)bp1;
        acc0 = __builtin_amdgcn_wmma_f32_16x16x4_f32(
            false, a, false, b0, (short)0, acc0, false, false);
        acc1 = __builtin_amdgcn_wmma_f32_16x16x4_f32(
            false, a, false, b1, (short)0, acc1, false, false);
        aptr += 4 * (size_t)HW;
        bp0  += 4;
        bp1  += 4;
    }

    // ---- Epilogue (branchless, select-based) ----
    // Lane owns out-channel o0 = n (and o1 = 16+n when n < 2) for 8 pixels.
    const float bo0 = bias[n];
    const float bo1 = bias[16 + (n & 1)];   // valid for n<2; harmless in-bounds read otherwise
    const int   a0  = n / 6;                // anchor index (uniform per lane)
    const int   ch0 = n % 6;                // channel within anchor
    const int   ch1 = (16 + n) % 6;         // o=16,17 -> anchor 2, ch 4/5 (sigmoid only)

    // Anchor scale via selects (exp(v)*anchor; the /stride*stride cancels)
    const float aw = (a0 == 0) ? 10.0f : ((a0 == 1) ? 16.0f : 33.0f);
    const float ah = (a0 == 0) ? 13.0f : ((a0 == 1) ? 30.0f : 23.0f);

    const bool isx = (ch0 == 0);
    const bool isy = (ch0 == 1);
    const bool isw = (ch0 == 2);
    const bool ish = (ch0 == 3);

    const float hf  = (float)h;
    const float wf0 = (float)(w0 + 8 * g);  // pixel x-coord for r = 0

    // Per-lane store bases; per-r offset is a constant (+6 floats per pixel).
    float* ob   = out + (size_t)b * (3 * HW * 6);
    const size_t prow = (size_t)(h * WIMG + w0 + 8 * g) * 6;
    float* st0 = ob + (size_t)a0 * (HW * 6) + prow + ch0;
    float* st1 = ob + (size_t)2  * (HW * 6) + prow + ch1;

#pragma unroll
    for (int r = 0; r < 8; ++r) {
        const float wf = wf0 + (float)r;

        const float v0  = acc0[r] + bo0;
        const float sig = sigmoidf(v0);
        const float ex  = __expf(v0);
        float r0;
        r0 = (isx | isy) ? (sig + (isx ? wf : hf)) * 8.0f
                         : (isw ? ex * aw : (ish ? ex * ah : sig));
        st0[r * 6] = r0;

        if (n < 2) {
            st1[r * 6] = sigmoidf(acc1[r] + bo1);   // ch 4/5: sigmoid only
        }
    }
}

extern "C" void kernel_launch(void* const* d_in, const int* in_sizes, int n_in,
                              void* d_out, int out_size, void* d_ws, size_t ws_size,
                              hipStream_t stream) {
    const float* z    = (const float*)d_in[0];   // z_f (B,C,H,W) fp32
    // d_in[1] = x_f: unused, matching reference
    const float* wgt  = (const float*)d_in[2];   // (18, 256)
    const float* bias = (const float*)d_in[3];   // (18,)
    float* out = (float*)d_out;                  // (16, 3*128*128, 6) fp32

    dim3 grid(2048);   // 16384 wave-tiles / 8 waves per block
    dim3 block(256);
    hipLaunchKernelGGL(yolo_wmma_kernel, grid, block, 0, stream, z, wgt, bias, out);
}